// VanillaGCNEncoder_56143812493986
// MI455X (gfx1250) — compile-verified
//
#include <hip/hip_runtime.h>
#include <cstdint>

typedef __attribute__((ext_vector_type(2))) float v2f;
typedef __attribute__((ext_vector_type(8))) float v8f;

// ---------------------------------------------------------------------------
// Degree / normalization kernels
// ---------------------------------------------------------------------------
__global__ __launch_bounds__(256) void deg_init_kernel(float* __restrict__ deg, int n) {
    int i = blockIdx.x * blockDim.x + threadIdx.x;
    if (i < n) deg[i] = 1.0f;  // self-loop contributes weight 1 to every node's degree
}

__global__ __launch_bounds__(256) void deg_accum_kernel(const long long* __restrict__ ei,
                                                        const float* __restrict__ ew,
                                                        float* __restrict__ deg, int E) {
    int e = blockIdx.x * blockDim.x + threadIdx.x;
    if (e < E) {
        long long d = ei[(size_t)E + e];  // dst row of edge_index
        unsafeAtomicAdd(&deg[d], ew[e]);
    }
}

__global__ __launch_bounds__(256) void dinv_kernel(const float* __restrict__ deg,
                                                   float* __restrict__ dinv, int n) {
    int i = blockIdx.x * blockDim.x + threadIdx.x;
    if (i < n) {
        float d = deg[i];
        dinv[i] = (d > 0.0f) ? rsqrtf(d) : 0.0f;
    }
}

// ---------------------------------------------------------------------------
// WMMA fp32 GEMM: out[M,NC] = A[M,K] @ W[K,NC]  (+ bias, tanh when FUSE)
// One wave per 16-row tile. K stepped by 4 with V_WMMA_F32_16X16X4_F32.
// A-frag (16x4 f32, 2 VGPR): lane L -> row L%16, K pair base (L/16)*2.
// B-frag (4x16 f32, 2 VGPR): lane L -> col L%16, K pair base (L/16)*2.
// C/D   (16x16 f32, 8 VGPR): vgpr v, lane L -> row v + 8*(L/16), col L%16.
// ---------------------------------------------------------------------------
template <int K, int NC, bool FUSE>
__global__ __launch_bounds__(256) void gemm_wmma_kernel(const float* __restrict__ A,
                                                        const float* __restrict__ W,
                                                        const float* __restrict__ bias,
                                                        float* __restrict__ out,
                                                        int numTiles) {
    const int wave = blockIdx.x * (blockDim.x >> 5) + (threadIdx.x >> 5);
    if (wave >= numTiles) return;  // wave-uniform: EXEC stays all-ones for WMMA

    const int lane    = threadIdx.x & 31;
    const int lrow    = lane & 15;
    const int koff    = (lane >> 4) << 1;   // 0 or 2
    const int rowBase = wave * 16;
    const int row     = rowBase + lrow;

    constexpr int NT = NC / 16;
    v8f acc[NT];
#pragma unroll
    for (int t = 0; t < NT; ++t)
#pragma unroll
        for (int i = 0; i < 8; ++i) acc[t][i] = 0.0f;

    for (int k = 0; k < K; k += 4) {
        // a.x, a.y are contiguous in memory -> single 8B load
        float2 av = *(const float2*)(A + (size_t)row * K + k + koff);
        v2f a; a.x = av.x; a.y = av.y;
#pragma unroll
        for (int t = 0; t < NT; ++t) {
            const int col = t * 16 + lrow;
            v2f b;
            b.x = W[(size_t)(k + koff) * NC + col];
            b.y = W[(size_t)(k + koff + 1) * NC + col];
            acc[t] = __builtin_amdgcn_wmma_f32_16x16x4_f32(
                false, a, false, b, (short)0, acc[t], false, false);
        }
    }

    const int rsel = (lane >> 4) * 8;
#pragma unroll
    for (int t = 0; t < NT; ++t) {
        const int col = t * 16 + lrow;
#pragma unroll
        for (int v = 0; v < 8; ++v) {
            const int r = rowBase + rsel + v;
            float val = acc[t][v];
            if (FUSE) val = tanhf(val + bias[col]);
            out[(size_t)r * NC + col] = val;
        }
    }
}

// ---------------------------------------------------------------------------
// Aggregation: a[i,:] = dinv[i]^2 * h[i,:]   (self-loop term, also zero-init)
// ---------------------------------------------------------------------------
__global__ __launch_bounds__(256) void agg_self_kernel(const float* __restrict__ h,
                                                       const float* __restrict__ dinv,
                                                       float* __restrict__ a, int total) {
    int gid = blockIdx.x * blockDim.x + threadIdx.x;
    if (gid < total) {
        int i = gid >> 5;  // width 32
        float di = dinv[i];
        a[gid] = di * di * h[gid];
    }
}

// One wave per edge, lane = feature: coalesced 128B gather + coalesced atomics.
__global__ __launch_bounds__(256) void agg_edge_kernel(const long long* __restrict__ ei,
                                                       const float* __restrict__ ew,
                                                       const float* __restrict__ dinv,
                                                       const float* __restrict__ h,
                                                       float* __restrict__ a, int E) {
    int gid = blockIdx.x * blockDim.x + threadIdx.x;
    int e = gid >> 5;
    int f = gid & 31;
    if (e < E) {
        long long s = ei[e];
        long long d = ei[(size_t)E + e];
        float norm = dinv[s] * ew[e] * dinv[d];
        unsafeAtomicAdd(&a[(size_t)d * 32 + f], norm * h[(size_t)s * 32 + f]);
    }
}

__global__ __launch_bounds__(256) void bias_relu_kernel(float* __restrict__ a,
                                                        const float* __restrict__ b,
                                                        int total) {
    int gid = blockIdx.x * blockDim.x + threadIdx.x;
    if (gid < total) {
        float v = a[gid] + b[gid & 31];
        a[gid] = v > 0.0f ? v : 0.0f;
    }
}

// ---------------------------------------------------------------------------
// Launch
// ---------------------------------------------------------------------------
extern "C" void kernel_launch(void* const* d_in, const int* in_sizes, int n_in,
                              void* d_out, int out_size, void* d_ws, size_t ws_size,
                              hipStream_t stream) {
    const float*     x  = (const float*)d_in[0];
    const long long* ei = (const long long*)d_in[1];  // int64 [2,E]
    const float*     ew = (const float*)d_in[2];
    const float*     W1 = (const float*)d_in[3];
    const float*     b1 = (const float*)d_in[4];
    const float*     W2 = (const float*)d_in[5];
    const float*     b2 = (const float*)d_in[6];
    const float*     W3 = (const float*)d_in[7];
    const float*     b3 = (const float*)d_in[8];
    float*           out = (float*)d_out;

    const int n = in_sizes[0] / 64;  // 100000
    const int E = in_sizes[2];       // 1600000

    // workspace layout (floats): deg[n] | dinv[n] | h[n*32] | a[n*32]
    float* deg  = (float*)d_ws;
    float* dinv = deg + n;
    float* h    = dinv + n;
    float* a    = h + (size_t)n * 32;

    auto cdiv = [](long long v, long long d) { return (int)((v + d - 1) / d); };

    // normalization
    deg_init_kernel<<<cdiv(n, 256), 256, 0, stream>>>(deg, n);
    deg_accum_kernel<<<cdiv(E, 256), 256, 0, stream>>>(ei, ew, deg, E);
    dinv_kernel<<<cdiv(n, 256), 256, 0, stream>>>(deg, dinv, n);

    const int tiles    = n / 16;            // 6250 (exact)
    const int gemmGrid = cdiv(tiles, 8);    // 8 waves / block
    const int nodeFeat = n * 32;
    const int edgeFeat = cdiv((long long)E * 32, 256);

    // layer 1: h = x @ W1 ; a = norm-agg(h) ; a = relu(a + b1)
    gemm_wmma_kernel<64, 32, false><<<gemmGrid, 256, 0, stream>>>(x, W1, nullptr, h, tiles);
    agg_self_kernel<<<cdiv(nodeFeat, 256), 256, 0, stream>>>(h, dinv, a, nodeFeat);
    agg_edge_kernel<<<edgeFeat, 256, 0, stream>>>(ei, ew, dinv, h, a, E);
    bias_relu_kernel<<<cdiv(nodeFeat, 256), 256, 0, stream>>>(a, b1, nodeFeat);

    // layer 2: h = a @ W2 ; a = norm-agg(h) ; a = relu(a + b2)
    gemm_wmma_kernel<32, 32, false><<<gemmGrid, 256, 0, stream>>>(a, W2, nullptr, h, tiles);
    agg_self_kernel<<<cdiv(nodeFeat, 256), 256, 0, stream>>>(h, dinv, a, nodeFeat);
    agg_edge_kernel<<<edgeFeat, 256, 0, stream>>>(ei, ew, dinv, h, a, E);
    bias_relu_kernel<<<cdiv(nodeFeat, 256), 256, 0, stream>>>(a, b2, nodeFeat);

    // layer 3: out = tanh(a @ W3 + b3)
    gemm_wmma_kernel<32, 16, true><<<gemmGrid, 256, 0, stream>>>(a, W3, b3, out, tiles);
}